// SemGraphConv_70523363000908
// MI455X (gfx1250) — compile-verified
//
#include <hip/hip_runtime.h>

#define N_FEAT 128           // feature width (f32)
#define OUT_ROW 256          // concat([h, h_N]) row width
#define WAVE 32              // gfx1250 is wave32-only
#define SCAN_BLK 256

// ---------------------------------------------------------------------------
// k0: zero per-node counters and the global segment cursor.
// ---------------------------------------------------------------------------
__global__ __launch_bounds__(256) void sgc_zero_kernel(
    int* __restrict__ counts, int* __restrict__ cursor, int nNodes)
{
    const int tid = blockIdx.x * blockDim.x + threadIdx.x;
    if (tid < nNodes) counts[tid] = 0;
    if (tid == 0) *cursor = 0;
}

// ---------------------------------------------------------------------------
// k1: in-degree histogram. One int atomic per edge (distributed, cheap).
// ---------------------------------------------------------------------------
__global__ __launch_bounds__(256) void sgc_hist_kernel(
    const int* __restrict__ dst, int* __restrict__ counts, int nEdges)
{
    const int e = blockIdx.x * blockDim.x + threadIdx.x;
    if (e >= nEdges) return;
    __hip_atomic_fetch_add(&counts[dst[e]], 1, __ATOMIC_RELAXED,
                           __HIP_MEMORY_SCOPE_AGENT);
}

// ---------------------------------------------------------------------------
// k2: per-block LDS inclusive scan of counts; one global-cursor atomic per
// block allocates the block's segment range. Segment placement order across
// blocks is irrelevant to the result (each node owns its own segment).
// ---------------------------------------------------------------------------
__global__ __launch_bounds__(SCAN_BLK) void sgc_offsets_kernel(
    const int* __restrict__ counts, int* __restrict__ off,
    int* __restrict__ cur, int* __restrict__ cursor, int nNodes)
{
    __shared__ int sdata[SCAN_BLK];
    __shared__ int sbase;

    const int t = threadIdx.x;
    const int n = blockIdx.x * SCAN_BLK + t;
    const int c = (n < nNodes) ? counts[n] : 0;

    sdata[t] = c;
    __syncthreads();
    // Hillis-Steele inclusive scan over 256 entries (8 steps, LDS).
    for (int o = 1; o < SCAN_BLK; o <<= 1) {
        int v = (t >= o) ? sdata[t - o] : 0;
        __syncthreads();
        sdata[t] += v;
        __syncthreads();
    }
    const int incl = sdata[t];
    if (t == 0) {
        sbase = __hip_atomic_fetch_add(cursor, sdata[SCAN_BLK - 1],
                                       __ATOMIC_RELAXED, __HIP_MEMORY_SCOPE_AGENT);
    }
    __syncthreads();

    if (n < nNodes) {
        const int o = sbase + incl - c;   // exclusive offset within global pool
        off[n] = o;
        cur[n] = o;                       // running cursor for the scatter pass
    }
}

// ---------------------------------------------------------------------------
// k3: scatter edge ids into their destination node's segment.
// ---------------------------------------------------------------------------
__global__ __launch_bounds__(256) void sgc_scatter_kernel(
    const int* __restrict__ dst, int* __restrict__ cur,
    int* __restrict__ eidx, int nEdges)
{
    const int e = blockIdx.x * blockDim.x + threadIdx.x;
    if (e >= nEdges) return;
    const int pos = __hip_atomic_fetch_add(&cur[dst[e]], 1, __ATOMIC_RELAXED,
                                           __HIP_MEMORY_SCOPE_AGENT);
    eidx[pos] = e;
}

// ---------------------------------------------------------------------------
// k4: one wave32 per node. Lane l owns float4 l of the 128-float row.
// Register accumulation over the node's edge list; 1-deep software pipeline
// with global_prefetch_b8 on the next source row (L2-resident gather).
// Writes BOTH output halves: [h, sum/max(deg,1)]. No atomics.
// ---------------------------------------------------------------------------
__global__ __launch_bounds__(256) void sgc_node_kernel(
    const float* __restrict__ h, const float* __restrict__ w,
    const int* __restrict__ src,
    const int* __restrict__ counts, const int* __restrict__ off,
    const int* __restrict__ eidx,
    float* __restrict__ out, int nNodes)
{
    const int lane = threadIdx.x & (WAVE - 1);
    const int n    = (blockIdx.x * blockDim.x + threadIdx.x) >> 5;
    if (n >= nNodes) return;

    const int deg  = counts[n];          // wave-uniform -> scalar loads
    const int base = off[n];

    float4 acc = make_float4(0.f, 0.f, 0.f, 0.f);

    int   s  = 0;
    float wt = 0.f;
    if (deg > 0) {
        const int e0 = eidx[base];
        s  = src[e0];
        wt = w[e0];
        __builtin_prefetch(h + (size_t)s * N_FEAT + lane * 4, 0, 3);
    }
    for (int k = 0; k < deg; ++k) {
        int   s_next  = s;
        float wt_next = wt;
        if (k + 1 < deg) {
            const int en = eidx[base + k + 1];
            s_next  = src[en];
            wt_next = w[en];
            // Warm L0/L2 for next iteration's 512B source row.
            __builtin_prefetch(h + (size_t)s_next * N_FEAT + lane * 4, 0, 3);
        }
        const float4 v = *((const float4*)(h + (size_t)s * N_FEAT) + lane);
        acc.x = fmaf(v.x, wt, acc.x);
        acc.y = fmaf(v.y, wt, acc.y);
        acc.z = fmaf(v.z, wt, acc.z);
        acc.w = fmaf(v.w, wt, acc.w);
        s  = s_next;
        wt = wt_next;
    }

    const float inv = 1.0f / fmaxf((float)deg, 1.0f);
    acc.x *= inv; acc.y *= inv; acc.z *= inv; acc.w *= inv;

    // Left half: copy of h[n]; right half: mean-aggregated neighborhood.
    const float4 hv = *((const float4*)(h + (size_t)n * N_FEAT) + lane);
    float4* outL = (float4*)(out + (size_t)n * OUT_ROW) + lane;
    float4* outR = (float4*)(out + (size_t)n * OUT_ROW + N_FEAT) + lane;
    *outL = hv;
    *outR = acc;
}

// ---------------------------------------------------------------------------
// Launch. Inputs: (h, w, src, dst). Output: [N, 256] f32.
// Workspace layout (ints): counts[N] | off[N] | cur[N] | cursor[1] | eidx[E]
// -> 3*N + 1 + E ints  (~3.85 MB for N=50k, E=800k).
// ---------------------------------------------------------------------------
extern "C" void kernel_launch(void* const* d_in, const int* in_sizes, int n_in,
                              void* d_out, int out_size, void* d_ws, size_t ws_size,
                              hipStream_t stream) {
    const float* h   = (const float*)d_in[0];
    const float* w   = (const float*)d_in[1];
    const int*   src = (const int*)d_in[2];
    const int*   dst = (const int*)d_in[3];
    float* out = (float*)d_out;

    const int nNodes = in_sizes[0] / N_FEAT;
    const int nEdges = in_sizes[1];

    int* counts = (int*)d_ws;
    int* off    = counts + nNodes;
    int* cur    = off + nNodes;
    int* cursor = cur + nNodes;
    int* eidx   = cursor + 1;

    const int nodeBlocks1 = (nNodes + 255) / 256;                 // 1 thread/node
    const int scanBlocks  = (nNodes + SCAN_BLK - 1) / SCAN_BLK;
    const int edgeBlocks  = (nEdges + 255) / 256;                 // 1 thread/edge
    const int waveBlocks  = (nNodes * WAVE + 255) / 256;          // 1 wave/node

    sgc_zero_kernel   <<<nodeBlocks1, 256, 0, stream>>>(counts, cursor, nNodes);
    sgc_hist_kernel   <<<edgeBlocks,  256, 0, stream>>>(dst, counts, nEdges);
    sgc_offsets_kernel<<<scanBlocks,  SCAN_BLK, 0, stream>>>(counts, off, cur,
                                                             cursor, nNodes);
    sgc_scatter_kernel<<<edgeBlocks,  256, 0, stream>>>(dst, cur, eidx, nEdges);
    sgc_node_kernel   <<<waveBlocks,  256, 0, stream>>>(h, w, src, counts, off,
                                                        eidx, out, nNodes);
}